// SS2D_block_23192823398803
// MI455X (gfx1250) — compile-verified
//
#include <hip/hip_runtime.h>
#include <hip/hip_bf16.h>
#include <math.h>

// SS2D block for MI455X (gfx1250, wave32).
// B=4, H=W=64 (L=4096), D=96, N=16, R=4, K=4.
//
// Pipeline:
//  1) proj_kernel   : x_dbl = W_proj[k] @ xs[b,k]  via V_WMMA_F32_16X16X4_F32
//                     (36x96 @ 96x4096), fused with delta = softplus(dtW @ dts + bias)
//                     (96x4 @ 4x4096, K=4 == native WMMA K). Writes delta, Bs, Cs.
//  2) scan_phase1   : per-chunk SSM transition operators (a,b) per (channel, state)
//  3) scan_phase2   : sequential compose across 16 chunks -> initial states
//  4) scan_phase3   : real scan per chunk with y = C.h + D*u  -> out_y
//  5) merge_kernel  : direction-dependent gather (flip / HW-transpose) + merge_w blend

#define L_ 4096
#define D_ 96
#define N_ 16
#define K_ 4
#define B_ 4
#define CC_ 36     // R + 2N
#define NCH 16     // number of scan chunks
#define CHUNK 256  // L_/NCH

typedef __attribute__((ext_vector_type(2))) float v2f;
typedef __attribute__((ext_vector_type(8))) float v8f;

__device__ __forceinline__ float softplus_f(float x) {
    return (x > 20.0f) ? x : log1pf(__expf(x));
}

// ---------------------------------------------------------------------------
// Kernel 1: projection GEMMs via fp32 WMMA.
// grid = (L_/16, K_, B_), block = 32 (one wave; EXEC all ones around WMMA).
// ---------------------------------------------------------------------------
__global__ __launch_bounds__(32) void proj_kernel(
    const float* __restrict__ x,      // (B,H,W,D)
    const float* __restrict__ xproj,  // (K, 36, 96)
    const float* __restrict__ dtw,    // (K, 96, 4)
    const float* __restrict__ dtb,    // (K, 96)
    float* __restrict__ delta,        // (B*K, L, D)
    float* __restrict__ Bsb,          // (B*K, L, N)
    float* __restrict__ Csb)          // (B*K, L, N)
{
    const int lane = threadIdx.x;
    const int half = lane >> 4;   // 0: K=0,1 ; 1: K=2,3
    const int n16  = lane & 15;
    const int k    = blockIdx.y;
    const int b    = blockIdx.z;
    const int col  = blockIdx.x * 16 + n16;   // l for this lane's B/D column
    const int bk   = b * K_ + k;

    // xs[b,k,d,col] gather offset (depends only on col,k,b; contiguous in d)
    int lp = (k >= 2) ? (L_ - 1 - col) : col;
    int lh = (k & 1) ? (lp & 63) : (lp >> 6);
    int lw = (k & 1) ? (lp >> 6) : (lp & 63);
    const float* xcol = x + (size_t)((b * 64 + lh) * 64 + lw) * D_;

    const float* Wk = xproj + (size_t)k * CC_ * D_;

    // Row pointers for the three A M-tiles; tile 2 rows >= 36 are clamped to a
    // legal address and zeroed by a multiply mask (keeps inner loop branch-free,
    // so EXEC stays all-ones straight through the WMMAs).
    const float* Wrow0 = Wk + (size_t)n16 * D_;
    const float* Wrow1 = Wk + (size_t)(16 + n16) * D_;
    const int   r2     = 32 + n16;
    const float m2     = (r2 < CC_) ? 1.0f : 0.0f;
    const float* Wrow2 = Wk + (size_t)((r2 < CC_) ? r2 : (CC_ - 1)) * D_;

    v8f acc0 = {}; v8f acc1 = {}; v8f acc2 = {};

    for (int dk = 0; dk < D_; dk += 4) {
        const int ca = dk + 2 * half;  // this lane-half's K columns ca, ca+1 (8B aligned)
        v2f bf = *(const v2f*)(xcol + ca);     // B[K][N] = xs[ca..ca+1][col]
        v2f a0 = *(const v2f*)(Wrow0 + ca);
        v2f a1 = *(const v2f*)(Wrow1 + ca);
        v2f a2 = *(const v2f*)(Wrow2 + ca);
        a2.x *= m2; a2.y *= m2;

        acc0 = __builtin_amdgcn_wmma_f32_16x16x4_f32(false, a0, false, bf, (short)0, acc0, false, false);
        acc1 = __builtin_amdgcn_wmma_f32_16x16x4_f32(false, a1, false, bf, (short)0, acc1, false, false);
        acc2 = __builtin_amdgcn_wmma_f32_16x16x4_f32(false, a2, false, bf, (short)0, acc2, false, false);
    }

    // Scatter x_dbl rows: c<4 -> dts (kept in regs), 4..19 -> Bs, 20..35 -> Cs.
    {
        const size_t rowbase = ((size_t)bk * L_ + col) * N_;
        #pragma unroll
        for (int r = 0; r < 8; ++r) {
            const int c0 = r + 8 * half;
            int c; float v;
            c = c0;      v = acc0[r];
            if (c >= 4)  Bsb[rowbase + (c - 4)] = v;
            c = 16 + c0; v = acc1[r];
            if (c < 20)  Bsb[rowbase + (c - 4)] = v;
            else         Csb[rowbase + (c - 20)] = v;
            c = 32 + c0; v = acc2[r];
            if (c < CC_) Csb[rowbase + (c - 20)] = v;
        }
    }

    // Build B fragment of second GEMM from dts rows (acc0 rows 0..3) via shuffles.
    const float t0 = __shfl(acc0[0], n16, 32);  // dts[0][n16]
    const float t1 = __shfl(acc0[1], n16, 32);  // dts[1][n16]
    const float t2 = __shfl(acc0[2], n16, 32);  // dts[2][n16]
    const float t3 = __shfl(acc0[3], n16, 32);  // dts[3][n16]
    v2f bf2; bf2.x = half ? t2 : t0; bf2.y = half ? t3 : t1;

    const float* Wd = dtw + (size_t)k * D_ * 4;
    const float* bb = dtb + (size_t)k * D_;
    float* dp = delta + ((size_t)bk * L_ + col) * D_;

    #pragma unroll
    for (int mt = 0; mt < 6; ++mt) {
        const int row = mt * 16 + n16;
        v2f a = *(const v2f*)(Wd + (size_t)row * 4 + 2 * half);
        v8f c = {};
        c = __builtin_amdgcn_wmma_f32_16x16x4_f32(false, a, false, bf2, (short)0, c, false, false);
        #pragma unroll
        for (int r = 0; r < 8; ++r) {
            const int d = mt * 16 + r + 8 * half;
            dp[d] = softplus_f(c[r] + bb[d]);
        }
    }
}

// ---------------------------------------------------------------------------
// Kernel 2: per-chunk transition operators.  h_out = a*h_in + b  (per state n)
// grid = (NCH, K_, B_), block = 96 (lane -> d).
// ---------------------------------------------------------------------------
__global__ __launch_bounds__(96) void scan_phase1(
    const float* __restrict__ x,
    const float* __restrict__ delta,   // (BK, L, D)
    const float* __restrict__ Bsb,     // (BK, L, N)
    const float* __restrict__ A_logs,  // (K*D, N)
    float* __restrict__ cA,            // (BK*D, NCH, N)
    float* __restrict__ cB)
{
    const int chunk = blockIdx.x, k = blockIdx.y, b = blockIdx.z;
    const int d = threadIdx.x;
    const int bk = b * K_ + k;

    float an[N_];
    {
        const float4* Aq = (const float4*)(A_logs + ((size_t)k * D_ + d) * N_);
        #pragma unroll
        for (int q = 0; q < 4; ++q) {
            float4 t = Aq[q];
            an[4*q+0] = -__expf(t.x); an[4*q+1] = -__expf(t.y);
            an[4*q+2] = -__expf(t.z); an[4*q+3] = -__expf(t.w);
        }
    }

    float pa[N_], pb[N_];
    #pragma unroll
    for (int n = 0; n < N_; ++n) { pa[n] = 1.0f; pb[n] = 0.0f; }

    const float* dpt = delta + (size_t)bk * L_ * D_;
    const float* bsp = Bsb + (size_t)bk * L_ * N_;
    const int l0 = chunk * CHUNK;

    for (int l = l0; l < l0 + CHUNK; ++l) {
        const float dl = dpt[(size_t)l * D_ + d];
        const int lp = (k >= 2) ? (L_ - 1 - l) : l;
        const int lh = (k & 1) ? (lp & 63) : (lp >> 6);
        const int lw = (k & 1) ? (lp >> 6) : (lp & 63);
        const float ul = x[(size_t)((b * 64 + lh) * 64 + lw) * D_ + d];
        const float du = dl * ul;
        // wave-uniform B vector: 4 x b128 loads
        float bvv[N_];
        {
            const float4* bq = (const float4*)(bsp + (size_t)l * N_);
            #pragma unroll
            for (int q = 0; q < 4; ++q) {
                float4 t = bq[q];
                bvv[4*q+0] = t.x; bvv[4*q+1] = t.y; bvv[4*q+2] = t.z; bvv[4*q+3] = t.w;
            }
        }
        #pragma unroll
        for (int n = 0; n < N_; ++n) {
            const float dA = __expf(dl * an[n]);
            pa[n] *= dA;
            pb[n] = fmaf(dA, pb[n], du * bvv[n]);
        }
    }

    const size_t base = (((size_t)bk * D_ + d) * NCH + chunk) * N_;
    #pragma unroll
    for (int n = 0; n < N_; ++n) { cA[base + n] = pa[n]; cB[base + n] = pb[n]; }
}

// ---------------------------------------------------------------------------
// Kernel 3: exclusive prefix of transition operators across chunks.
// one thread per (channel, state): 1536*16 = 24576 threads.
// ---------------------------------------------------------------------------
__global__ void scan_phase2(const float* __restrict__ cA,
                            const float* __restrict__ cB,
                            float* __restrict__ hinit)
{
    const int idx = blockIdx.x * blockDim.x + threadIdx.x;
    if (idx >= B_ * K_ * D_ * N_) return;
    const int n = idx & (N_ - 1);
    const int c = idx >> 4;
    float h = 0.0f;
    const size_t base = (size_t)c * NCH * N_ + n;
    for (int j = 0; j < NCH; ++j) {
        hinit[base + (size_t)j * N_] = h;
        h = fmaf(cA[base + (size_t)j * N_], h, cB[base + (size_t)j * N_]);
    }
}

// ---------------------------------------------------------------------------
// Kernel 4: real scan per chunk with correct initial state; y = C.h + Ds*u.
// ---------------------------------------------------------------------------
__global__ __launch_bounds__(96) void scan_phase3(
    const float* __restrict__ x,
    const float* __restrict__ delta,
    const float* __restrict__ Bsb,
    const float* __restrict__ Csb,
    const float* __restrict__ A_logs,
    const float* __restrict__ Ds,      // (K*D,)
    const float* __restrict__ hinit,   // (BK*D, NCH, N)
    float* __restrict__ out_y)         // (BK, L, D)
{
    const int chunk = blockIdx.x, k = blockIdx.y, b = blockIdx.z;
    const int d = threadIdx.x;
    const int bk = b * K_ + k;

    float an[N_];
    {
        const float4* Aq = (const float4*)(A_logs + ((size_t)k * D_ + d) * N_);
        #pragma unroll
        for (int q = 0; q < 4; ++q) {
            float4 t = Aq[q];
            an[4*q+0] = -__expf(t.x); an[4*q+1] = -__expf(t.y);
            an[4*q+2] = -__expf(t.z); an[4*q+3] = -__expf(t.w);
        }
    }

    float h[N_];
    {
        const float4* hq = (const float4*)(hinit + (((size_t)bk * D_ + d) * NCH + chunk) * N_);
        #pragma unroll
        for (int q = 0; q < 4; ++q) {
            float4 t = hq[q];
            h[4*q+0] = t.x; h[4*q+1] = t.y; h[4*q+2] = t.z; h[4*q+3] = t.w;
        }
    }

    const float dsc = Ds[(size_t)k * D_ + d];
    const float* dpt = delta + (size_t)bk * L_ * D_;
    const float* bsp = Bsb + (size_t)bk * L_ * N_;
    const float* csp = Csb + (size_t)bk * L_ * N_;
    float* yp = out_y + (size_t)bk * L_ * D_;
    const int l0 = chunk * CHUNK;

    for (int l = l0; l < l0 + CHUNK; ++l) {
        const float dl = dpt[(size_t)l * D_ + d];
        const int lp = (k >= 2) ? (L_ - 1 - l) : l;
        const int lh = (k & 1) ? (lp & 63) : (lp >> 6);
        const int lw = (k & 1) ? (lp >> 6) : (lp & 63);
        const float ul = x[(size_t)((b * 64 + lh) * 64 + lw) * D_ + d];
        const float du = dl * ul;
        // wave-uniform B and C vectors: 4 x b128 loads each
        float bvv[N_], cvv[N_];
        {
            const float4* bq = (const float4*)(bsp + (size_t)l * N_);
            const float4* cq = (const float4*)(csp + (size_t)l * N_);
            #pragma unroll
            for (int q = 0; q < 4; ++q) {
                float4 tb = bq[q];
                bvv[4*q+0] = tb.x; bvv[4*q+1] = tb.y; bvv[4*q+2] = tb.z; bvv[4*q+3] = tb.w;
                float4 tc = cq[q];
                cvv[4*q+0] = tc.x; cvv[4*q+1] = tc.y; cvv[4*q+2] = tc.z; cvv[4*q+3] = tc.w;
            }
        }
        float y = 0.0f;
        #pragma unroll
        for (int n = 0; n < N_; ++n) {
            const float dA = __expf(dl * an[n]);
            h[n] = fmaf(dA, h[n], du * bvv[n]);
            y = fmaf(h[n], cvv[n], y);
        }
        yp[(size_t)l * D_ + d] = y + dsc * ul;    // coalesced in d
    }
}

// ---------------------------------------------------------------------------
// Kernel 5: direction merge.  out(b,h,w,d) = sum_k mw[k]*ystack_k + mb
// ystack: [k0, flip(k2), HWtranspose(k1), flip+HWtranspose(k3)]
// ---------------------------------------------------------------------------
__global__ void merge_kernel(const float* __restrict__ out_y,
                             const float* __restrict__ mw,
                             const float* __restrict__ mb,
                             float* __restrict__ out)
{
    const int idx = blockIdx.x * blockDim.x + threadIdx.x;
    if (idx >= B_ * L_ * D_) return;
    const int d = idx % D_;
    const int l = (idx / D_) & (L_ - 1);
    const int b = idx / (D_ * L_);
    const int hh = l >> 6, ww = l & 63;
    const int lt = ww * 64 + hh;

    const size_t base = (size_t)b * K_;
    const float y0 = out_y[((base + 0) * L_ + l) * D_ + d];
    const float y1 = out_y[((base + 2) * L_ + (L_ - 1 - l)) * D_ + d];
    const float y2 = out_y[((base + 1) * L_ + lt) * D_ + d];
    const float y3 = out_y[((base + 3) * L_ + (L_ - 1 - lt)) * D_ + d];

    out[idx] = mw[0] * y0 + mw[1] * y1 + mw[2] * y2 + mw[3] * y3 + mb[0];
}

// ---------------------------------------------------------------------------
extern "C" void kernel_launch(void* const* d_in, const int* in_sizes, int n_in,
                              void* d_out, int out_size, void* d_ws, size_t ws_size,
                              hipStream_t stream) {
    (void)in_sizes; (void)n_in; (void)out_size; (void)ws_size;
    const float* x     = (const float*)d_in[0];
    const float* xproj = (const float*)d_in[1];
    const float* dtw   = (const float*)d_in[2];
    const float* dtb   = (const float*)d_in[3];
    const float* Alogs = (const float*)d_in[4];
    const float* Ds    = (const float*)d_in[5];
    const float* mw    = (const float*)d_in[6];
    const float* mb    = (const float*)d_in[7];
    float* out = (float*)d_out;

    float* wsf = (float*)d_ws;
    size_t off = 0;
    float* delta = wsf + off; off += (size_t)B_ * K_ * L_ * D_;   // 6.29M
    float* Bsb   = wsf + off; off += (size_t)B_ * K_ * L_ * N_;   // 1.05M
    float* Csb   = wsf + off; off += (size_t)B_ * K_ * L_ * N_;   // 1.05M
    float* cA    = wsf + off; off += (size_t)B_ * K_ * D_ * NCH * N_;
    float* cB    = wsf + off; off += (size_t)B_ * K_ * D_ * NCH * N_;
    float* hinit = wsf + off; off += (size_t)B_ * K_ * D_ * NCH * N_;
    float* outy  = wsf + off; off += (size_t)B_ * K_ * L_ * D_;   // 6.29M

    proj_kernel<<<dim3(L_ / 16, K_, B_), 32, 0, stream>>>(
        x, xproj, dtw, dtb, delta, Bsb, Csb);

    scan_phase1<<<dim3(NCH, K_, B_), 96, 0, stream>>>(
        x, delta, Bsb, Alogs, cA, cB);

    scan_phase2<<<(B_ * K_ * D_ * N_ + 255) / 256, 256, 0, stream>>>(
        cA, cB, hinit);

    scan_phase3<<<dim3(NCH, K_, B_), 96, 0, stream>>>(
        x, delta, Bsb, Csb, Alogs, Ds, hinit, outy);

    merge_kernel<<<(B_ * L_ * D_ + 255) / 256, 256, 0, stream>>>(
        outy, mw, mb, out);
}